// ChamferLoss_19181323944568
// MI455X (gfx1250) — compile-verified
//
#include <hip/hip_runtime.h>
#include <math.h>

typedef __attribute__((ext_vector_type(2))) float v2f;
typedef __attribute__((ext_vector_type(8))) float v8f;

#define WAVES            8                      // 256 threads = 8 wave32
#define STRIPS           2                      // 16-row strips per wave
#define ROWS_PER_BLOCK   (WAVES * STRIPS * 16)  // 256 rows of P per block

__global__ void chamfer_zero_kernel(float* out) { out[0] = 0.0f; }

// 2 column-tiles (32 columns) of WMMA + min3 accumulation for all strips.
__device__ __forceinline__ void chamfer_tiles(const v2f afrag[STRIPS],
                                              v2f bfrag0, v2f bfrag1,
                                              v8f rowmin[STRIPS]) {
    #pragma unroll
    for (int s = 0; s < STRIPS; ++s) {
        v8f c0 = {};
        v8f c1 = {};
        c0 = __builtin_amdgcn_wmma_f32_16x16x4_f32(
                 false, afrag[s], false, bfrag0, (short)0, c0, false, false);
        c1 = __builtin_amdgcn_wmma_f32_16x16x4_f32(
                 false, afrag[s], false, bfrag1, (short)0, c1, false, false);
        #pragma unroll
        for (int v = 0; v < 8; ++v)
            rowmin[s][v] = fminf(fminf(c0[v], c1[v]), rowmin[s][v]);
    }
}

// Both directions of Chamfer in one grid. dir 0: P=pred vs T=target,
// dir 1: roles swapped. For each point in P, min distance over all of T;
// scale * sum of mins accumulated atomically into *out.
//
// Algebra: ||p - t||^2 = ||p||^2 + (||t||^2 - 2 p.t); the parenthesized part
// is one WMMA:
//   A row m = (-2px, -2py, -2pz, 1)          (K padded to 4)
//   B col j = (tx,   ty,   tz,   ||t||^2)
//   D[m][j] = -2 p.t + ||t||^2
// ||p||^2 is row-constant => hoisted past the min over j; sqrt is monotone
// => applied only to the final 2*B*N minima.
//
// LDS staging: per column j we pre-pack exactly what each half-wave's B
// fragment needs (16x4 f32 B layout: lanes 0-15 hold (K0,K1), lanes 16-31
// hold (K2,K3), N = lane % 16):
//   smem[0..2n)  : bf0[j] = (tx, ty)
//   smem[2n..4n) : bf1[j] = (tz, ||t||^2)
// Inner loop = 2x ds_load_2addr_b64 (pipelined one iteration ahead) +
// 8x v_wmma + 32x v_min3 per 64 columns.
__global__ __launch_bounds__(256)
void chamfer_both_kernel(const float* __restrict__ pred,
                         const float* __restrict__ targ,
                         float* __restrict__ out,
                         int n, int blocksPerDir, float scale) {
    extern __shared__ float smem[];   // 4*n floats = 16n bytes (128KB @ n=8192)

    const int dir = blockIdx.x / blocksPerDir;
    const int bid = blockIdx.x % blocksPerDir;
    const float* Pm = dir ? targ : pred;
    const float* Tm = dir ? pred : targ;

    const int blocksPerBatch = n / ROWS_PER_BLOCK;
    const int b    = bid / blocksPerBatch;
    const int rb   = bid % blocksPerBatch;
    const int tid  = threadIdx.x;
    const int wave = tid >> 5;
    const int lane = tid & 31;
    const int half = lane >> 4;   // 0: lanes 0-15, 1: lanes 16-31
    const int r    = lane & 15;

    const float* p = Pm + (size_t)b * 3 * n;
    const float* t = Tm + (size_t)b * 3 * n;

    // ---- Stage packed B fragments for the whole T cloud into LDS.
    for (int j = tid; j < n; j += 256) {
        const float tx = t[j];
        const float ty = t[n + j];
        const float tz = t[2 * n + j];
        const float t2 = tx * tx + ty * ty + tz * tz;
        *(v2f*)(smem + 2 * j)         = (v2f){tx, ty};
        *(v2f*)(smem + 2 * n + 2 * j) = (v2f){tz, t2};
    }

    // ---- A fragments (16x4 f32 A layout: lanes 0-15 hold (K0,K1),
    // lanes 16-31 hold (K2,K3), M = lane % 16).
    const int rowBase = rb * ROWS_PER_BLOCK + wave * STRIPS * 16;
    v2f afrag[STRIPS];
    #pragma unroll
    for (int s = 0; s < STRIPS; ++s) {
        const int m = rowBase + s * 16 + r;
        const float px = p[m];
        const float py = p[n + m];
        const float pz = p[2 * n + m];
        afrag[s].x = half ? (-2.0f * pz) : (-2.0f * px);
        afrag[s].y = half ? 1.0f         : (-2.0f * py);
    }

    v8f rowmin[STRIPS];
    #pragma unroll
    for (int s = 0; s < STRIPS; ++s)
        #pragma unroll
        for (int v = 0; v < 8; ++v)
            rowmin[s][v] = 3.4e38f;

    __syncthreads();

    // ---- Sweep all columns, 32 per iteration, with the B-fragment LDS loads
    // software-pipelined one iteration ahead so s_wait_dscnt never stalls on
    // a just-issued load.
    const float* bptr = smem + (half ? 2 * n : 0) + 2 * r;
    v2f bf0 = *(const v2f*)(bptr);
    v2f bf1 = *(const v2f*)(bptr + 32);
    #pragma unroll 2
    for (int colBase = 0; colBase < n - 32; colBase += 32) {
        const v2f nbf0 = *(const v2f*)(bptr + 2 * colBase + 64);
        const v2f nbf1 = *(const v2f*)(bptr + 2 * colBase + 96);
        chamfer_tiles(afrag, bf0, bf1, rowmin);
        bf0 = nbf0;
        bf1 = nbf1;
    }
    chamfer_tiles(afrag, bf0, bf1, rowmin);   // peeled last 32 columns

    // ---- Row-min across the 16 lanes holding different columns.
    // xor masks 1,2,4,8 keep the shuffle inside each 16-lane half (wave32).
    #pragma unroll
    for (int s = 0; s < STRIPS; ++s) {
        #pragma unroll
        for (int v = 0; v < 8; ++v) {
            float x = rowmin[s][v];
            #pragma unroll
            for (int mask = 1; mask <= 8; mask <<= 1)
                x = fminf(x, __shfl_xor(x, mask, 32));
            rowmin[s][v] = x;
        }
    }

    // Lanes r==0 of each half hold the 8 row minima of that half. Re-add the
    // hoisted ||p||^2 per row, clamp, sqrt, sum, and accumulate.
    if (r == 0) {
        float localSum = 0.0f;
        #pragma unroll
        for (int s = 0; s < STRIPS; ++s) {
            #pragma unroll
            for (int v = 0; v < 8; ++v) {
                const int m = rowBase + s * 16 + v + 8 * half;
                const float px = p[m];
                const float py = p[n + m];
                const float pz = p[2 * n + m];
                const float p2 = px * px + py * py + pz * pz;
                localSum += sqrtf(fmaxf(p2 + rowmin[s][v], 0.0f));
            }
        }
        atomicAdd(out, localSum * scale);
    }
}

extern "C" void kernel_launch(void* const* d_in, const int* in_sizes, int n_in,
                              void* d_out, int out_size, void* d_ws, size_t ws_size,
                              hipStream_t stream) {
    const float* pred   = (const float*)d_in[0];  // [B, 3, N] f32
    const float* target = (const float*)d_in[1];  // [B, 3, N] f32
    float* out = (float*)d_out;                   // scalar f32

    const int B = 4;
    const int n = in_sizes[0] / (3 * B);          // 8192
    const float scale = 1.0f / (float)(B * n);

    chamfer_zero_kernel<<<1, 1, 0, stream>>>(out);

    const int blocksPerDir = B * (n / ROWS_PER_BLOCK);     // 128
    const size_t ldsBytes = (size_t)4 * n * sizeof(float); // 128KB
    chamfer_both_kernel<<<2 * blocksPerDir, 256, ldsBytes, stream>>>(
        pred, target, out, n, blocksPerDir, scale);
}